// VectorQuantizer_86921548137095
// MI455X (gfx1250) — compile-verified
//
#include <hip/hip_runtime.h>
#include <math.h>

typedef __attribute__((ext_vector_type(2))) float    v2f;
typedef __attribute__((ext_vector_type(8))) float    v8f;
typedef __attribute__((ext_vector_type(4))) unsigned u32x4;
typedef __attribute__((ext_vector_type(8))) unsigned u32x8;

#define DIMD   256
#define KCODES 1024
#define NROWS  65536          // B*H*W
#define NEL    16777216       // B*D*H*W

#define ROWS_PER_WG 128
#define CHUNK       64        // codes per TDM chunk
#define NCHUNK      (KCODES / CHUNK)
#define BSTRIDE     260       // padded floats per codebook row in LDS (TDM pad: 4 dwords / 256)
#define A_BYTES     (8 * 64 * 32 * 2 * 4)      // 131072: [wave][t][lane][2] f32
#define B_BYTES     (CHUNK * BSTRIDE * 4)      // 66560
#define SMEM_BYTES  (A_BYTES + 2 * B_BYTES)    // 264192 < 320KB

// workspace layout (bytes)
#define WS_LOSS   0           // double
#define WS_COUNTS 16          // 1024 u32
#define WS_E2     4128        // 1024 f32
#define WS_IDX    8224        // 65536 i32

// ---------------------------------------------------------------------------
// TDM: async load one 64x256 f32 tile of the codebook into LDS, with padding
// so each 256-dword row is followed by 4 pad dwords (row stride = 260 floats).
// 2-group descriptor (2D tensor), per CDNA5 ISA.
// ---------------------------------------------------------------------------
__device__ __forceinline__ void tdm_load_chunk(const float* cb, unsigned lds_byte_off, int chunk) {
  unsigned long long ga =
      (unsigned long long)(size_t)cb + (unsigned long long)chunk * (CHUNK * DIMD * 4);
  u32x4 g0;
  g0[0] = 1u;                                            // count=1, user desc
  g0[1] = lds_byte_off;                                  // lds_addr
  g0[2] = (unsigned)(ga & 0xFFFFFFFFu);                  // global_addr[31:0]
  g0[3] = (unsigned)((ga >> 32) & 0x01FFFFFFu) | (2u << 30); // global_addr[56:32] | type=2
  u32x8 g1;
  // data_size=2 (4B) | pad_enable | pad_interval=7 (256 dw) | pad_amount=3 (4 dw)
  g1[0] = (2u << 16) | (1u << 20) | (7u << 22) | (3u << 25);
  g1[1] = ((unsigned)DIMD & 0xFFFFu) << 16;              // tensor_dim0 lo16 (=256)
  g1[2] = ((unsigned)KCODES & 0xFFFFu) << 16;            // dim0 hi=0 | tensor_dim1 lo16 (=1024)
  g1[3] = ((unsigned)DIMD & 0xFFFFu) << 16;              // dim1 hi=0 | tile_dim0 (=256)
  g1[4] = (unsigned)CHUNK;                               // tile_dim1=64 | tile_dim2=0
  g1[5] = (unsigned)DIMD;                                // tensor_dim0_stride lo32 (=256)
  g1[6] = 0u;
  g1[7] = 0u;
  asm volatile("tensor_load_to_lds %0, %1" :: "s"(g0), "s"(g1) : "memory");
}

// ---------------------------------------------------------------------------
// Kernel: codebook row squared norms
// ---------------------------------------------------------------------------
__global__ void vq_norms_kernel(const float* __restrict__ cb, float* __restrict__ e2) {
  const int lane = threadIdx.x & 31;
  const int w    = threadIdx.x >> 5;
  const int k    = blockIdx.x * 8 + w;
  const float* row = cb + (size_t)k * DIMD;
  float s = 0.f;
  #pragma unroll
  for (int d = 0; d < DIMD; d += 32) { float v = row[d + lane]; s = fmaf(v, v, s); }
  #pragma unroll
  for (int off = 16; off; off >>= 1) s += __shfl_xor(s, off, 32);
  if (lane == 0) e2[k] = s;
}

// ---------------------------------------------------------------------------
// Kernel: fp32 WMMA distance GEMM + running argmin
// 8 waves/WG; wave owns 16 rows; loop K in 64-code LDS chunks (TDM double-buffered)
// ---------------------------------------------------------------------------
__global__ __launch_bounds__(256, 1)
void vq_argmin_kernel(const float* __restrict__ latents,
                      const float* __restrict__ codebook,
                      const float* __restrict__ e2,
                      int* __restrict__ idx) {
  extern __shared__ char smem[];
  float* Alds = (float*)smem;
  const int tid  = threadIdx.x;
  const int wg   = blockIdx.x;
  const int lane = tid & 31;
  const int w    = tid >> 5;

  const unsigned lds_base = (unsigned)(size_t)(&smem[0]);

  // prefetch chunk 0 while we stage A (distinct LDS regions)
  if (w == 0) tdm_load_chunk(codebook, lds_base + A_BYTES, 0);

  // stage this WG's 128 rows x 256 dims of latents into WMMA-A layout.
  // flat row n = b*1024 + hw ; element at latents[((b*256+d)<<10) + hw]
  const int n0  = wg * ROWS_PER_WG;
  const int b   = n0 >> 10;
  const int hw0 = n0 & 1023;
  const float* latB = latents + ((size_t)b * DIMD) * 1024 + hw0;
  #pragma unroll 4
  for (int it = 0; it < (ROWS_PER_WG * DIMD) / 256; ++it) {
    const int linear = it * 256 + tid;
    const int d = linear >> 7;          // dim
    const int m = linear & 127;         // local row
    const float v = latB[(size_t)d * 1024 + m];
    const int wv = m >> 4, l = m & 15, t = d >> 2, ks = d & 3;
    const int ln = l + ((ks >> 1) << 4);
    Alds[(((wv * 64 + t) * 32) + ln) * 2 + (ks & 1)] = v;
  }
  __syncthreads();

  float minv[8]; int mini[8];
  #pragma unroll
  for (int j = 0; j < 8; ++j) { minv[j] = 3.4e38f; mini[j] = 0; }

  const int colLane = lane & 15;
  const int khalf   = lane >> 4;
  const float* Abase = Alds + ((size_t)(w * 64) * 32 + lane) * 2;

  for (int c = 0; c < NCHUNK; ++c) {
    __builtin_amdgcn_s_wait_tensorcnt(0);   // wave 0: chunk c landed; others: no-op
    __syncthreads();                        // all waves see chunk c; prev compute done
    if (w == 0 && (c + 1) < NCHUNK)
      tdm_load_chunk(codebook, lds_base + A_BYTES + (unsigned)(((c + 1) & 1) * B_BYTES), c + 1);

    const float* Bbuf = (const float*)(smem + A_BYTES + (c & 1) * B_BYTES);
    const float* Bp0 = Bbuf + (0 * 16 + colLane) * BSTRIDE + 2 * khalf;
    const float* Bp1 = Bbuf + (1 * 16 + colLane) * BSTRIDE + 2 * khalf;
    const float* Bp2 = Bbuf + (2 * 16 + colLane) * BSTRIDE + 2 * khalf;
    const float* Bp3 = Bbuf + (3 * 16 + colLane) * BSTRIDE + 2 * khalf;

    v8f acc0 = {}, acc1 = {}, acc2 = {}, acc3 = {};
    #pragma unroll 4
    for (int t = 0; t < 64; ++t) {
      v2f a  = *(const v2f*)(Abase + (size_t)t * 64);
      v2f b0 = *(const v2f*)(Bp0 + 4 * t);
      v2f b1 = *(const v2f*)(Bp1 + 4 * t);
      v2f b2 = *(const v2f*)(Bp2 + 4 * t);
      v2f b3 = *(const v2f*)(Bp3 + 4 * t);
      acc0 = __builtin_amdgcn_wmma_f32_16x16x4_f32(false, a, false, b0, (short)0, acc0, false, false);
      acc1 = __builtin_amdgcn_wmma_f32_16x16x4_f32(false, a, false, b1, (short)0, acc1, false, false);
      acc2 = __builtin_amdgcn_wmma_f32_16x16x4_f32(false, a, false, b2, (short)0, acc2, false, false);
      acc3 = __builtin_amdgcn_wmma_f32_16x16x4_f32(false, a, false, b3, (short)0, acc3, false, false);
    }

    // fold dist = ||e||^2 - 2*dot, track argmin (||x||^2 is row-constant: irrelevant)
    v8f accs[4] = {acc0, acc1, acc2, acc3};
    #pragma unroll
    for (int ct = 0; ct < 4; ++ct) {
      const int col = c * CHUNK + ct * 16 + colLane;
      const float ecol = e2[col];
      #pragma unroll
      for (int j = 0; j < 8; ++j) {
        float dist = fmaf(-2.0f, accs[ct][j], ecol);
        if (dist < minv[j]) { minv[j] = dist; mini[j] = col; }
      }
    }
  }

  // cross-lane argmin within each 16-lane half (C layout: VGPR j -> M=j (lanes 0-15), M=j+8 (16-31))
  #pragma unroll
  for (int off = 1; off < 16; off <<= 1) {
    #pragma unroll
    for (int j = 0; j < 8; ++j) {
      float ov = __shfl_xor(minv[j], off, 32);
      int   oi = __shfl_xor(mini[j], off, 32);
      if (ov < minv[j] || (ov == minv[j] && oi < mini[j])) { minv[j] = ov; mini[j] = oi; }
    }
  }
  if (colLane == 0) {
    const int rbase = wg * ROWS_PER_WG + w * 16 + khalf * 8;
    #pragma unroll
    for (int j = 0; j < 8; ++j) idx[rbase + j] = mini[j];
  }
}

// ---------------------------------------------------------------------------
// Kernel: code usage histogram (LDS-local then global atomics)
// ---------------------------------------------------------------------------
__global__ void vq_hist_kernel(const int* __restrict__ idx, unsigned* __restrict__ counts) {
  __shared__ unsigned hc[KCODES];
  const int tid = threadIdx.x;
  for (int i = tid; i < KCODES; i += 256) hc[i] = 0u;
  __syncthreads();
  const int base = blockIdx.x * 1024;
  for (int i = tid; i < 1024; i += 256) atomicAdd(&hc[idx[base + i]], 1u);
  __syncthreads();
  for (int i = tid; i < KCODES; i += 256) { unsigned c = hc[i]; if (c) atomicAdd(&counts[i], c); }
}

// ---------------------------------------------------------------------------
// Kernel: gather quantized output (STE forward == quantized) + loss partial
// ---------------------------------------------------------------------------
__global__ __launch_bounds__(256)
void vq_gather_loss_kernel(const float* __restrict__ latents, const float* __restrict__ cb,
                           const int* __restrict__ idx, float* __restrict__ out,
                           double* __restrict__ lossAcc) {
  const size_t id = (size_t)blockIdx.x * 256 + threadIdx.x;
  const int bq = (int)(id >> 18);        // 256*1024 elements per batch image
  const int r  = (int)(id & 262143);
  const int d  = r >> 10;
  const int hw = r & 1023;
  const int n  = (bq << 10) | hw;
  const int k  = idx[n];
  const float q = cb[(size_t)k * DIMD + d];
  const float x = latents[id];
  out[id] = q;
  const float diff = q - x;
  float s = diff * diff;
  #pragma unroll
  for (int off = 16; off; off >>= 1) s += __shfl_xor(s, off, 32);
  __shared__ float wsum[8];
  const int lane = threadIdx.x & 31, w = threadIdx.x >> 5;
  if (lane == 0) wsum[w] = s;
  __syncthreads();
  if (threadIdx.x == 0) {
    float t = 0.f;
    #pragma unroll
    for (int i = 0; i < 8; ++i) t += wsum[i];
    atomicAdd(lossAcc, (double)t);
  }
}

// ---------------------------------------------------------------------------
// Kernel: scalars -> vq_loss, perplexity
// ---------------------------------------------------------------------------
__global__ void vq_finalize_kernel(const unsigned* __restrict__ counts,
                                   const double* __restrict__ lossAcc,
                                   float* __restrict__ out) {
  __shared__ float sh[256];
  const int tid = threadIdx.x;
  float s = 0.f;
  for (int i = tid; i < KCODES; i += 256) {
    float p = (float)counts[i] * (1.0f / (float)NROWS);
    s += p * logf(p + 1e-10f);
  }
  sh[tid] = s; __syncthreads();
  for (int st = 128; st; st >>= 1) { if (tid < st) sh[tid] += sh[tid + st]; __syncthreads(); }
  if (tid == 0) {
    out[(size_t)NEL]     = (float)(1.25 * (*lossAcc) / (double)NEL);  // e_loss + 0.25*q_loss
    out[(size_t)NEL + 1] = expf(-sh[0]);
  }
}

// ---------------------------------------------------------------------------
extern "C" void kernel_launch(void* const* d_in, const int* in_sizes, int n_in,
                              void* d_out, int out_size, void* d_ws, size_t ws_size,
                              hipStream_t stream) {
  const float* latents  = (const float*)d_in[0];
  const float* codebook = (const float*)d_in[1];
  float* out = (float*)d_out;
  char*  ws  = (char*)d_ws;
  double*   lossAcc = (double*)(ws + WS_LOSS);
  unsigned* counts  = (unsigned*)(ws + WS_COUNTS);
  float*    e2      = (float*)(ws + WS_E2);
  int*      idx     = (int*)(ws + WS_IDX);

  hipMemsetAsync(ws, 0, WS_COUNTS + KCODES * 4, stream);   // zero loss + counts
  vq_norms_kernel<<<KCODES / 8, 256, 0, stream>>>(codebook, e2);
  vq_argmin_kernel<<<NROWS / ROWS_PER_WG, 256, SMEM_BYTES, stream>>>(latents, codebook, e2, idx);
  vq_hist_kernel<<<NROWS / 1024, 256, 0, stream>>>(idx, counts);
  vq_gather_loss_kernel<<<NEL / 256, 256, 0, stream>>>(latents, codebook, idx, out, lossAcc);
  vq_finalize_kernel<<<1, 256, 0, stream>>>(counts, lossAcc, out);
}